// M8_81071802679814
// MI455X (gfx1250) — compile-verified
//
#include <hip/hip_runtime.h>

#define N_NODES 100000
#define N_EDGES 1600000
#define F_IN    64
#define H_DIM   48
#define K_CHEB  5
#define C_OUT   2
#define BN_EPS  1e-5f
#define LRELU   0.01f

typedef float v2f __attribute__((ext_vector_type(2)));
typedef float v8f __attribute__((ext_vector_type(8)));

// ---------------------------------------------------------------- utilities

__global__ void zero_kernel(float* __restrict__ p, int n) {
    int i = blockIdx.x * 256 + threadIdx.x;
    if (i < n) p[i] = 0.0f;
}

__global__ void scale_copy_kernel(float* __restrict__ out, const float* __restrict__ in,
                                  float alpha, int n) {
    int i = blockIdx.x * 256 + threadIdx.x;
    if (i < n) out[i] = alpha * in[i];
}

// ------------------------------------------------------- graph normalization

__global__ void deg_kernel(const int* __restrict__ src, float* __restrict__ deg) {
    int e = blockIdx.x * 256 + threadIdx.x;
    if (e < N_EDGES) atomicAdd(&deg[src[e]], 1.0f);
}

__global__ void dinv_kernel(const float* __restrict__ deg, float* __restrict__ dinv) {
    int i = blockIdx.x * 256 + threadIdx.x;
    if (i < N_NODES) {
        float d = deg[i];
        dinv[i] = (d > 0.0f) ? rsqrtf(fmaxf(d, 1e-12f)) : 0.0f;
    }
}

__global__ void norm_kernel(const int* __restrict__ src, const int* __restrict__ dst,
                            const float* __restrict__ dinv, float* __restrict__ nrm) {
    int e = blockIdx.x * 256 + threadIdx.x;
    if (e < N_EDGES) nrm[e] = -dinv[src[e]] * dinv[dst[e]];
}

// -------------------------------------------------- sparse propagation (L·x)
// tout[dst] += scale * nrm[e] * tin[src] ; edge-parallel, float4 per thread.
// Consecutive threads share an edge -> index/norm loads hit L0/L2; feature
// matrices (<=25.6 MB) are L2-resident on the 192 MB L2.

template<int FDIM>
__global__ void prop_kernel(const int* __restrict__ src, const int* __restrict__ dst,
                            const float* __restrict__ nrm,
                            const float* __restrict__ tin, float* __restrict__ tout,
                            float scale) {
    const int FQ = FDIM / 4;
    int idx = blockIdx.x * 256 + threadIdx.x;
    if (idx >= N_EDGES * FQ) return;
    int e = idx / FQ;
    int f = (idx - e * FQ) * 4;
    float w = scale * nrm[e];
    const float4 v = *(const float4*)(tin + (size_t)src[e] * FDIM + f);
    float* o = tout + (size_t)dst[e] * FDIM + f;
    atomicAdd(o + 0, w * v.x);
    atomicAdd(o + 1, w * v.y);
    atomicAdd(o + 2, w * v.z);
    atomicAdd(o + 3, w * v.w);
}

// ------------------------------------------------- dense GEMM via fp32 WMMA
// acc[N x 48] += X[N x FDIM] @ W[FDIM x 48] with V_WMMA_F32_16X16X4_F32.
// One wave owns one 16x16 output tile; K loop chains the accumulator.
// A frag : lane m = row, half selects K pair {0,1}/{2,3}  -> v2f
// B frag : lane m = col, half selects K pair              -> v2f
// C/D    : VGPR v -> row (v + 8*half), lane -> col        -> v8f

template<int FDIM>
__global__ void gemm_wmma_kernel(const float* __restrict__ X, const float* __restrict__ W,
                                 float* __restrict__ acc) {
    const int TN = H_DIM / 16;                       // 3 column tiles
    int wave = (blockIdx.x * blockDim.x + threadIdx.x) >> 5;
    if (wave >= (N_NODES / 16) * TN) return;         // wave-uniform: EXEC stays all-1s
    int lane = threadIdx.x & 31;
    int tm = wave / TN;
    int tn = wave - tm * TN;
    int m    = lane & 15;
    int half = lane >> 4;

    const float* xr = X + (size_t)(tm * 16 + m) * FDIM;
    const float* wc = W + tn * 16 + m;

    v8f c = {0.f, 0.f, 0.f, 0.f, 0.f, 0.f, 0.f, 0.f};
#pragma unroll
    for (int k = 0; k < FDIM; k += 4) {
        int ka = k + 2 * half;
        v2f a, b;
        a.x = xr[ka];
        a.y = xr[ka + 1];
        b.x = wc[(size_t)ka * H_DIM];
        b.y = wc[(size_t)(ka + 1) * H_DIM];
        c = __builtin_amdgcn_wmma_f32_16x16x4_f32(false, a, false, b,
                                                  (short)0, c, false, false);
    }

    float* o = acc + (size_t)(tm * 16 + 8 * half) * H_DIM + tn * 16 + m;
#pragma unroll
    for (int v = 0; v < 8; ++v) o[(size_t)v * H_DIM] += c[v];
}

// ------------------------------------------------------------ BN + LeakyReLU
// Conv bias cancels under train-mode BN (per-column mean subtraction), so it
// is omitted for layers followed by BN.

__global__ void colstats_kernel(const float* __restrict__ X, float* __restrict__ red) {
    __shared__ float s[2 * H_DIM];
    int tid = threadIdx.x;
    if (tid < 2 * H_DIM) s[tid] = 0.0f;
    __syncthreads();
    int idx = blockIdx.x * 256 + tid;
    if (idx < N_NODES * H_DIM) {
        float v = X[idx];
        int c = idx % H_DIM;
        atomicAdd(&s[c], v);
        atomicAdd(&s[H_DIM + c], v * v);
    }
    __syncthreads();
    if (tid < 2 * H_DIM) atomicAdd(&red[tid], s[tid]);
}

__global__ void bn_finalize_kernel(const float* __restrict__ red, const float* __restrict__ g,
                                   const float* __restrict__ bt, float* __restrict__ sc) {
    int c = threadIdx.x;
    if (c < H_DIM) {
        const float invN = 1.0f / (float)N_NODES;
        float mu  = red[c] * invN;
        float var = red[H_DIM + c] * invN - mu * mu;
        float a   = g[c] * rsqrtf(var + BN_EPS);
        sc[c]          = a;
        sc[H_DIM + c]  = bt[c] - mu * a;
    }
}

__global__ void bn_apply_kernel(const float* __restrict__ X, const float* __restrict__ sc,
                                float* __restrict__ Y) {
    int idx = blockIdx.x * 256 + threadIdx.x;
    if (idx < N_NODES * H_DIM) {
        int c = idx % H_DIM;
        float y = fmaf(X[idx], sc[c], sc[H_DIM + c]);
        Y[idx] = (y >= 0.0f) ? y : LRELU * y;
    }
}

// ------------------------------------------------- fused (h + b3) @ Wf + bf

__global__ void final_kernel(const float* __restrict__ Hm, const float* __restrict__ b3,
                             const float* __restrict__ Wf, const float* __restrict__ bf,
                             float* __restrict__ out) {
    int i = blockIdx.x * 256 + threadIdx.x;
    if (i < N_NODES) {
        const float* hr = Hm + (size_t)i * H_DIM;
        float a0 = bf[0], a1 = bf[1];
#pragma unroll
        for (int h = 0; h < H_DIM; ++h) {
            float v = hr[h] + b3[h];
            a0 = fmaf(v, Wf[h * C_OUT + 0], a0);
            a1 = fmaf(v, Wf[h * C_OUT + 1], a1);
        }
        out[i * C_OUT + 0] = a0;
        out[i * C_OUT + 1] = a1;
    }
}

// ---------------------------------------------------------------- host side

static inline int cdiv(long long a, int b) { return (int)((a + b - 1) / b); }

template<int FDIM>
static void cheb_layer(const float* X, const float* W,
                       const int* src, const int* dst, const float* nrm,
                       float* txa, float* txb, float* acc, hipStream_t stream) {
    const int NF = N_NODES * FDIM;
    const int PROP_BLK = cdiv((long long)N_EDGES * (FDIM / 4), 256);
    const int NEL_BLK  = cdiv(NF, 256);
    const int GEMM_BLK = cdiv((N_NODES / 16) * (H_DIM / 16), 8);  // 8 waves/block
    const size_t WS = (size_t)FDIM * H_DIM;                        // W slice stride

    zero_kernel<<<cdiv(N_NODES * H_DIM, 256), 256, 0, stream>>>(acc, N_NODES * H_DIM);
    // k = 0: acc += X @ W0
    gemm_wmma_kernel<FDIM><<<GEMM_BLK, 256, 0, stream>>>(X, W, acc);
    // k = 1: Tx1 = L x
    zero_kernel<<<NEL_BLK, 256, 0, stream>>>(txa, NF);
    prop_kernel<FDIM><<<PROP_BLK, 256, 0, stream>>>(src, dst, nrm, X, txa, 1.0f);
    gemm_wmma_kernel<FDIM><<<GEMM_BLK, 256, 0, stream>>>(txa, W + 1 * WS, acc);
    // k = 2: Tx2 = 2 L Tx1 - Tx0   (init -Tx0, scatter with scale 2)
    scale_copy_kernel<<<NEL_BLK, 256, 0, stream>>>(txb, X, -1.0f, NF);
    prop_kernel<FDIM><<<PROP_BLK, 256, 0, stream>>>(src, dst, nrm, txa, txb, 2.0f);
    gemm_wmma_kernel<FDIM><<<GEMM_BLK, 256, 0, stream>>>(txb, W + 2 * WS, acc);
    // k = 3: Tx3 = 2 L Tx2 - Tx1   (negate txa in place, reuse)
    scale_copy_kernel<<<NEL_BLK, 256, 0, stream>>>(txa, txa, -1.0f, NF);
    prop_kernel<FDIM><<<PROP_BLK, 256, 0, stream>>>(src, dst, nrm, txb, txa, 2.0f);
    gemm_wmma_kernel<FDIM><<<GEMM_BLK, 256, 0, stream>>>(txa, W + 3 * WS, acc);
    // k = 4: Tx4 = 2 L Tx3 - Tx2
    scale_copy_kernel<<<NEL_BLK, 256, 0, stream>>>(txb, txb, -1.0f, NF);
    prop_kernel<FDIM><<<PROP_BLK, 256, 0, stream>>>(src, dst, nrm, txa, txb, 2.0f);
    gemm_wmma_kernel<FDIM><<<GEMM_BLK, 256, 0, stream>>>(txb, W + 4 * WS, acc);
}

static void bn_lrelu(const float* acc, const float* g, const float* bt,
                     float* red, float* y, hipStream_t stream) {
    zero_kernel<<<1, 256, 0, stream>>>(red, 2 * H_DIM);
    colstats_kernel<<<cdiv(N_NODES * H_DIM, 256), 256, 0, stream>>>(acc, red);
    bn_finalize_kernel<<<1, 64, 0, stream>>>(red, g, bt, red + 2 * H_DIM);
    bn_apply_kernel<<<cdiv(N_NODES * H_DIM, 256), 256, 0, stream>>>(acc, red + 2 * H_DIM, y);
}

extern "C" void kernel_launch(void* const* d_in, const int* in_sizes, int n_in,
                              void* d_out, int out_size, void* d_ws, size_t ws_size,
                              hipStream_t stream) {
    const float* x   = (const float*)d_in[0];
    const int*   ei  = (const int*)d_in[1];
    const int*   src = ei;
    const int*   dst = ei + N_EDGES;
    const float* W1  = (const float*)d_in[2];
    const float* W2  = (const float*)d_in[4];
    const float* W3  = (const float*)d_in[6];
    const float* b3  = (const float*)d_in[7];
    const float* g1  = (const float*)d_in[8];
    const float* bt1 = (const float*)d_in[9];
    const float* g2  = (const float*)d_in[10];
    const float* bt2 = (const float*)d_in[11];
    const float* Wf  = (const float*)d_in[12];
    const float* bf  = (const float*)d_in[13];

    float* ws   = (float*)d_ws;
    size_t off  = 0;
    float* deg  = ws + off; off += N_NODES;
    float* dinv = ws + off; off += N_NODES;
    float* nrm  = ws + off; off += N_EDGES;
    float* txa  = ws + off; off += (size_t)N_NODES * F_IN;
    float* txb  = ws + off; off += (size_t)N_NODES * F_IN;
    float* acc  = ws + off; off += (size_t)N_NODES * H_DIM;
    float* hbuf = ws + off; off += (size_t)N_NODES * H_DIM;
    float* red  = ws + off; off += 256;

    // sym-norm:  nrm[e] = -d^{-1/2}[src] * d^{-1/2}[dst]
    zero_kernel<<<cdiv(N_NODES, 256), 256, 0, stream>>>(deg, N_NODES);
    deg_kernel<<<cdiv(N_EDGES, 256), 256, 0, stream>>>(src, deg);
    dinv_kernel<<<cdiv(N_NODES, 256), 256, 0, stream>>>(deg, dinv);
    norm_kernel<<<cdiv(N_EDGES, 256), 256, 0, stream>>>(src, dst, dinv, nrm);

    // layer 1 (F=64 -> 48) + BN/LReLU
    cheb_layer<F_IN>(x, W1, src, dst, nrm, txa, txb, acc, stream);
    bn_lrelu(acc, g1, bt1, red, hbuf, stream);

    // layer 2 (48 -> 48) + BN/LReLU
    cheb_layer<H_DIM>(hbuf, W2, src, dst, nrm, txa, txb, acc, stream);
    bn_lrelu(acc, g2, bt2, red, hbuf, stream);

    // layer 3 (48 -> 48), no BN; fused classifier
    cheb_layer<H_DIM>(hbuf, W3, src, dst, nrm, txa, txb, acc, stream);
    final_kernel<<<cdiv(N_NODES, 256), 256, 0, stream>>>(acc, b3, Wf, bf, (float*)d_out);
}